// SelectiveSSM_85598698209292
// MI455X (gfx1250) — compile-verified
//
#include <hip/hip_runtime.h>
#include <hip/hip_bf16.h>

#define D_MODEL 1024
#define D_STATE 16
#define D_CONV  4
#define DT_RANK 64
#define D_INNER 2048
#define BATCH   2
#define SEQ     1024
#define M_TOTAL (BATCH * SEQ)           // 2048 token rows
#define XDBL_W  (DT_RANK + 2 * D_STATE) // 96

// Use CDNA5 async global->LDS loads (ASYNCcnt) for A-tile staging.
#define USE_ASYNC_LDS 1

#define LDS_STRIDE 36                   // f32 stride: 16B-aligned rows, conflict-free

typedef __attribute__((ext_vector_type(2))) float v2f;
typedef __attribute__((ext_vector_type(4))) float v4f;
typedef __attribute__((ext_vector_type(8))) float v8f;

__device__ __forceinline__ float silu_f(float x) { return x / (1.0f + __expf(-x)); }
__device__ __forceinline__ float softplus_f(float x) {
    return (x > 20.0f) ? x : log1pf(__expf(x));
}

#if USE_ASYNC_LDS
__device__ __forceinline__ void async_load_b128(uint32_t lds_addr, const float* gptr) {
    uint64_t ga = (uint64_t)(uintptr_t)gptr;
    asm volatile("global_load_async_to_lds_b128 %0, %1, off"
                 :: "v"(lds_addr), "v"(ga) : "memory");
}
__device__ __forceinline__ void wait_async_le2() {
    asm volatile("s_wait_asynccnt 0x2" ::: "memory");
}
__device__ __forceinline__ void wait_async_le0() {
    asm volatile("s_wait_asynccnt 0x0" ::: "memory");
}
#endif

// ---------------------------------------------------------------------------
// f32 WMMA GEMM:  out[m][n] = epilogue( sum_k A[m][k] * W[n][k] )
//   A: [M x lda] row-major activations; W: [N x K] row-major weights.
// Block: 256 threads = 8 waves. Block tile 64(M) x 128(N); each wave owns
// 4 M-subtiles x 16(N) -> 4 accumulators, so each B fragment feeds 4 WMMAs.
// A K-chunk (64x32 f32) is staged in LDS; with USE_ASYNC_LDS the next chunk
// streams in via global_load_async_to_lds_b128 (double-buffered, ASYNCcnt).
// MODE 0: in-proj split (n < D_INNER -> out0 raw ; else -> out1 silu)
// MODE 1: plain
// MODE 2: softplus(acc + bias[n])
// ---------------------------------------------------------------------------
template <int MODE>
__global__ __launch_bounds__(256) void gemm_wmma_f32(
    const float* __restrict__ A, int lda,
    const float* __restrict__ W,
    const float* __restrict__ bias,
    float* __restrict__ out0, float* __restrict__ out1,
    int M, int N, int K)
{
    __shared__ float aTile[2][64 * LDS_STRIDE];

    const int tid  = threadIdx.x;
    const int wave = tid >> 5;
    const int lane = tid & 31;
    const int half = lane >> 4;   // 0: lanes 0-15, 1: lanes 16-31
    const int nl   = lane & 15;

    const int mBase = blockIdx.y * 64;
    const int nBase = blockIdx.x * 128 + wave * 16;
    const bool active = (nBase < N);          // all N are multiples of 16

    v8f c0 = {}, c1 = {}, c2 = {}, c3 = {};
    const float* wRow = W + (size_t)(nBase + nl) * (size_t)K;

    // cooperative A staging coords: 256 threads x 2 b128 = 64 rows x 32 cols
    const int sr = tid >> 3;            // 0..31 (rows sr and sr+32)
    const int sc = (tid & 7) << 2;      // 0,4,...,28

#if USE_ASYNC_LDS
    // ---- double-buffered async pipeline -----------------------------------
    {   // prologue: stage chunk 0 into buffer 0
        const float* g = A + (size_t)(mBase + sr) * (size_t)lda + sc;
        uint32_t l = (uint32_t)(uintptr_t)(&aTile[0][sr * LDS_STRIDE + sc]);
        async_load_b128(l, g);
        async_load_b128(l + 32 * LDS_STRIDE * 4, g + (size_t)32 * lda);
    }
    int p = 0;
    for (int kBase = 0; kBase < K; kBase += 32) {
        const bool haveNext = (kBase + 32) < K;
        if (haveNext) {
            const float* g = A + (size_t)(mBase + sr) * (size_t)lda + (kBase + 32) + sc;
            uint32_t l = (uint32_t)(uintptr_t)(&aTile[1 - p][sr * LDS_STRIDE + sc]);
            async_load_b128(l, g);
            async_load_b128(l + 32 * LDS_STRIDE * 4, g + (size_t)32 * lda);
            wait_async_le2();       // chunk-p data (issued earlier) is complete
        } else {
            wait_async_le0();
        }
        __syncthreads();            // chunk-p visible from all threads

        if (active) {
#pragma unroll
            for (int s = 0; s < 8; ++s) {
                const int ko = s * 4 + half * 2;
                v2f b = *(const v2f*)(wRow + kBase + ko);
                v2f a0 = *(const v2f*)(&aTile[p][(nl)      * LDS_STRIDE + ko]);
                v2f a1 = *(const v2f*)(&aTile[p][(nl + 16) * LDS_STRIDE + ko]);
                v2f a2 = *(const v2f*)(&aTile[p][(nl + 32) * LDS_STRIDE + ko]);
                v2f a3 = *(const v2f*)(&aTile[p][(nl + 48) * LDS_STRIDE + ko]);
                c0 = __builtin_amdgcn_wmma_f32_16x16x4_f32(false, a0, false, b, (short)0, c0, false, false);
                c1 = __builtin_amdgcn_wmma_f32_16x16x4_f32(false, a1, false, b, (short)0, c1, false, false);
                c2 = __builtin_amdgcn_wmma_f32_16x16x4_f32(false, a2, false, b, (short)0, c2, false, false);
                c3 = __builtin_amdgcn_wmma_f32_16x16x4_f32(false, a3, false, b, (short)0, c3, false, false);
            }
        }
        __syncthreads();            // all waves done reading buf p before overwrite
        p ^= 1;
    }
#else
    // ---- fallback: synchronous staging ------------------------------------
    for (int kBase = 0; kBase < K; kBase += 32) {
        {
            const float* g = A + (size_t)(mBase + sr) * (size_t)lda + kBase + sc;
            v4f v0 = *(const v4f*)(g);
            v4f v1 = *(const v4f*)(g + (size_t)32 * lda);
            *(v4f*)(&aTile[0][sr * LDS_STRIDE + sc])        = v0;
            *(v4f*)(&aTile[0][(sr + 32) * LDS_STRIDE + sc]) = v1;
        }
        __syncthreads();
        if (active) {
#pragma unroll
            for (int s = 0; s < 8; ++s) {
                const int ko = s * 4 + half * 2;
                v2f b = *(const v2f*)(wRow + kBase + ko);
                v2f a0 = *(const v2f*)(&aTile[0][(nl)      * LDS_STRIDE + ko]);
                v2f a1 = *(const v2f*)(&aTile[0][(nl + 16) * LDS_STRIDE + ko]);
                v2f a2 = *(const v2f*)(&aTile[0][(nl + 32) * LDS_STRIDE + ko]);
                v2f a3 = *(const v2f*)(&aTile[0][(nl + 48) * LDS_STRIDE + ko]);
                c0 = __builtin_amdgcn_wmma_f32_16x16x4_f32(false, a0, false, b, (short)0, c0, false, false);
                c1 = __builtin_amdgcn_wmma_f32_16x16x4_f32(false, a1, false, b, (short)0, c1, false, false);
                c2 = __builtin_amdgcn_wmma_f32_16x16x4_f32(false, a2, false, b, (short)0, c2, false, false);
                c3 = __builtin_amdgcn_wmma_f32_16x16x4_f32(false, a3, false, b, (short)0, c3, false, false);
            }
        }
        __syncthreads();
    }
#endif

    if (active) {
        const int n = nBase + nl;
#pragma unroll
        for (int t = 0; t < 4; ++t) {
            v8f c = (t == 0) ? c0 : (t == 1) ? c1 : (t == 2) ? c2 : c3;
#pragma unroll
            for (int r = 0; r < 8; ++r) {
                const int m = mBase + t * 16 + r + half * 8;
                float v = c[r];
                if (MODE == 0) {
                    if (n < D_INNER)
                        out0[(size_t)m * D_INNER + n] = v;
                    else
                        out1[(size_t)m * D_INNER + (n - D_INNER)] = silu_f(v);
                } else if (MODE == 1) {
                    out0[(size_t)m * (size_t)N + n] = v;
                } else {
                    out0[(size_t)m * (size_t)N + n] = softplus_f(v + bias[n]);
                }
            }
        }
    }
}

// ---------------------------------------------------------------------------
// Causal depthwise conv1d (pad left D_CONV-1) + SiLU. One thread per (m, d).
// ---------------------------------------------------------------------------
__global__ __launch_bounds__(256) void conv_silu_kernel(
    const float* __restrict__ xs,     // [M_TOTAL, D_INNER]
    const float* __restrict__ Wc,     // [D_INNER, D_CONV]
    const float* __restrict__ bc,     // [D_INNER]
    float* __restrict__ xc)           // [M_TOTAL, D_INNER]
{
    const int idx = blockIdx.x * 256 + threadIdx.x;
    const int d = idx & (D_INNER - 1);
    const int m = idx >> 11;           // / D_INNER
    const int l = m & (SEQ - 1);       // position within batch

    float acc = bc[d];
#pragma unroll
    for (int k = 0; k < D_CONV; ++k) {
        const int ls = l + k - (D_CONV - 1);
        if (ls >= 0)
            acc = fmaf(Wc[d * D_CONV + k],
                       xs[(size_t)(m + k - (D_CONV - 1)) * D_INNER + d], acc);
    }
    xc[(size_t)m * D_INNER + d] = silu_f(acc);
}

// ---------------------------------------------------------------------------
// Selective scan: one thread per (batch, channel). 16 states + 16 A values in
// registers; B/C per-step loads are wave-uniform (L2 broadcast).
// Fused epilogue: yg = (y + x*D) * silu(z)   [z already silu'ed]
// ---------------------------------------------------------------------------
__global__ __launch_bounds__(256) void selective_scan_kernel(
    const float* __restrict__ xc,
    const float* __restrict__ dt,
    const float* __restrict__ xdbl,   // [M_TOTAL, XDBL_W] (B at +64, C at +80)
    const float* __restrict__ A_log,  // [D_INNER, D_STATE]
    const float* __restrict__ Dv,
    const float* __restrict__ z,
    float* __restrict__ yg)
{
    const int idx = blockIdx.x * 256 + threadIdx.x;
    const int d = idx & (D_INNER - 1);
    const int b = idx >> 11;

    float Arow[D_STATE], h[D_STATE];
#pragma unroll
    for (int n = 0; n < D_STATE; ++n) {
        Arow[n] = -__expf(A_log[d * D_STATE + n]);
        h[n] = 0.0f;
    }
    const float Dd = Dv[d];

    for (int l = 0; l < SEQ; ++l) {
        const int m = b * SEQ + l;
        const float xv  = xc[(size_t)m * D_INNER + d];
        const float dtv = dt[(size_t)m * D_INNER + d];
        const float* bcrow = xdbl + (size_t)m * XDBL_W + DT_RANK;

        float y = 0.0f;
#pragma unroll
        for (int n = 0; n < D_STATE; ++n) {
            const float dA = __expf(dtv * Arow[n]);     // v_exp_f32
            const float Bn = bcrow[n];
            const float Cn = bcrow[D_STATE + n];
            h[n] = fmaf(dA, h[n], dtv * Bn * xv);
            y = fmaf(h[n], Cn, y);
        }
        yg[(size_t)m * D_INNER + d] = (y + xv * Dd) * z[(size_t)m * D_INNER + d];
    }
}

// ---------------------------------------------------------------------------
extern "C" void kernel_launch(void* const* d_in, const int* in_sizes, int n_in,
                              void* d_out, int out_size, void* d_ws, size_t ws_size,
                              hipStream_t stream)
{
    const float* x      = (const float*)d_in[0];
    const float* W_in   = (const float*)d_in[1];
    const float* W_conv = (const float*)d_in[2];
    const float* b_conv = (const float*)d_in[3];
    const float* W_x    = (const float*)d_in[4];
    const float* W_dt   = (const float*)d_in[5];
    const float* b_dt   = (const float*)d_in[6];
    const float* A_log  = (const float*)d_in[7];
    const float* Dvec   = (const float*)d_in[8];
    const float* W_out  = (const float*)d_in[9];
    float* out = (float*)d_out;

    const size_t SZ = (size_t)M_TOTAL * D_INNER;      // 4M floats
    float* ws   = (float*)d_ws;
    float* xs   = ws;                                  // x_ssm (pre-conv)
    float* zbuf = xs   + SZ;                           // silu(z)
    float* xc   = zbuf + SZ;                           // conv+silu output
    float* dtb  = xc   + SZ;                           // dt (softplus)
    float* yg   = dtb  + SZ;                           // gated scan output
    float* xdbl = yg   + SZ;                           // [M_TOTAL, 96]

    // 1) in-proj GEMM + split/silu: M=2048 N=4096 K=1024
    {
        dim3 grid((2 * D_INNER) / 128, M_TOTAL / 64);
        gemm_wmma_f32<0><<<grid, 256, 0, stream>>>(
            x, D_MODEL, W_in, nullptr, xs, zbuf, M_TOTAL, 2 * D_INNER, D_MODEL);
    }
    // 2) causal depthwise conv + silu
    {
        dim3 grid((unsigned)(SZ / 256));
        conv_silu_kernel<<<grid, 256, 0, stream>>>(xs, W_conv, b_conv, xc);
    }
    // 3) x_dbl = x_conv @ W_x^T : M=2048 N=96 K=2048
    {
        dim3 grid(1, M_TOTAL / 64);
        gemm_wmma_f32<1><<<grid, 256, 0, stream>>>(
            xc, D_INNER, W_x, nullptr, xdbl, nullptr, M_TOTAL, XDBL_W, D_INNER);
    }
    // 4) dt = softplus(dt_low @ W_dt^T + b_dt) : M=2048 N=2048 K=64 (lda=96)
    {
        dim3 grid(D_INNER / 128, M_TOTAL / 64);
        gemm_wmma_f32<2><<<grid, 256, 0, stream>>>(
            xdbl, XDBL_W, W_dt, b_dt, dtb, nullptr, M_TOTAL, D_INNER, DT_RANK);
    }
    // 5) selective scan + gating
    {
        dim3 grid((BATCH * D_INNER) / 256);
        selective_scan_kernel<<<grid, 256, 0, stream>>>(
            xc, dtb, xdbl, A_log, Dvec, zbuf, yg);
    }
    // 6) out-proj GEMM: M=2048 N=1024 K=2048 -> d_out
    {
        dim3 grid(D_MODEL / 128, M_TOTAL / 64);
        gemm_wmma_f32<1><<<grid, 256, 0, stream>>>(
            yg, D_INNER, W_out, nullptr, out, nullptr, M_TOTAL, D_MODEL, D_INNER);
    }
}